// PointSampling_37306085933345
// MI455X (gfx1250) — compile-verified
//
#include <hip/hip_runtime.h>
#include <cstdint>
#include <cstddef>

#define BATCH 8
#define NPTS  16384   // N, power of two (1<<14)
#define NCH   128     // C (1<<7)
#define MOUT  8192    // npoint = N*0.5 (1<<13)
#define TPB   1024    // 32 waves of 32
#define PPT   (NPTS / TPB)  // 16 points per thread

// ---------------------------------------------------------------------------
// wave32 max-reduction of a packed u64 key via lane shuffles
// ---------------------------------------------------------------------------
__device__ __forceinline__ unsigned long long wave_max_u64(unsigned long long key) {
#pragma unroll
  for (int off = 16; off > 0; off >>= 1) {
    unsigned lo = __shfl_xor((unsigned)(key & 0xFFFFFFFFull), off, 32);
    unsigned hi = __shfl_xor((unsigned)(key >> 32), off, 32);
    unsigned long long other = ((unsigned long long)hi << 32) | lo;
    if (other > key) key = other;
  }
  return key;
}

// ---------------------------------------------------------------------------
// Furthest point sampling: one workgroup per batch. xyz tile lives in LDS
// (SoA, 192KB of the WGP's 320KB), per-point running distances live in VGPRs.
// Staging HBM->LDS uses CDNA5 async global->LDS loads with a per-lane
// transposed LDS address (AoS -> SoA in one pass, tracked by ASYNCcnt).
// ---------------------------------------------------------------------------
__global__ __launch_bounds__(TPB) void fps_kernel(const float* __restrict__ xyz,
                                                  int* __restrict__ idx_out) {
  __shared__ float sx[NPTS];
  __shared__ float sy[NPTS];
  __shared__ float sz[NPTS];
  __shared__ unsigned long long wavekeys[2][TPB / 32];

  const int b   = blockIdx.x;
  const int tid = threadIdx.x;
  const float* base = xyz + (size_t)b * NPTS * 3;

  // ---- Stage xyz (AoS in HBM) into LDS as SoA via async global->LDS ----
#pragma unroll
  for (int k = 0; k < PPT; ++k) {
    int i = tid + k * TPB;
    unsigned lx = (unsigned)(uintptr_t)(&sx[i]);
    unsigned ly = (unsigned)(uintptr_t)(&sy[i]);
    unsigned lz = (unsigned)(uintptr_t)(&sz[i]);
    unsigned long long g = (unsigned long long)(uintptr_t)(base + (size_t)i * 3);
    asm volatile("global_load_async_to_lds_b32 %0, %1, off"
                 :: "v"(lx), "v"(g) : "memory");
    asm volatile("global_load_async_to_lds_b32 %0, %1, off offset:4"
                 :: "v"(ly), "v"(g) : "memory");
    asm volatile("global_load_async_to_lds_b32 %0, %1, off offset:8"
                 :: "v"(lz), "v"(g) : "memory");
  }
  asm volatile("s_wait_asynccnt 0x0" ::: "memory");
  __syncthreads();

  // ---- running min-distances resident in registers ----
  float dist[PPT];
#pragma unroll
  for (int k = 0; k < PPT; ++k) dist[k] = 1e10f;

  int cur = 0;  // first emitted index is 0 (matches lax.scan carry init)
  for (int it = 0; it < MOUT; ++it) {
    if (tid == 0) idx_out[(size_t)b * MOUT + it] = cur;

    // centroid broadcast from LDS (uniform address)
    float cx = sx[cur], cy = sy[cur], cz = sz[cur];

    float bestd = -1.0f;
    int   bestn = 0;
#pragma unroll
    for (int k = 0; k < PPT; ++k) {
      int n = tid + k * TPB;
      float dx = sx[n] - cx;
      float dy = sy[n] - cy;
      float dz = sz[n] - cz;
      // unfused, left-to-right like the reference's elementwise square + sum
      float d  = __fadd_rn(__fadd_rn(__fmul_rn(dx, dx), __fmul_rn(dy, dy)),
                           __fmul_rn(dz, dz));
      float dd = fminf(dist[k], d);
      dist[k] = dd;
      if (dd > bestd) { bestd = dd; bestn = n; }  // strict '>' => first max
    }

    // pack (dist, index): non-negative float bits are order-preserving;
    // ~index makes max() prefer the SMALLEST index on ties (jnp.argmax).
    unsigned long long key =
        ((unsigned long long)__float_as_uint(bestd) << 32) |
        (unsigned)(~(unsigned)bestn);
    key = wave_max_u64(key);

    const int slot = it & 1;                 // ping-pong: 1 barrier / iter
    if ((tid & 31) == 0) wavekeys[slot][tid >> 5] = key;
    __syncthreads();

    // every wave redundantly reduces the 32 per-wave keys -> no 2nd barrier
    unsigned long long k2 = wavekeys[slot][tid & 31];
    k2  = wave_max_u64(k2);
    cur = (int)(~(unsigned)(k2 & 0xFFFFFFFFull));
  }
}

// ---------------------------------------------------------------------------
// new_feats[b, c, m] = feats[b, c, idx[b, m]]   (coalesced writes over m;
// scattered reads stay inside 64KB rows, whole feats tensor fits in L2)
// ---------------------------------------------------------------------------
__global__ __launch_bounds__(256) void gather_feats_kernel(
    const float* __restrict__ feats, const int* __restrict__ idx,
    float* __restrict__ out) {
  size_t t  = (size_t)blockIdx.x * 256 + threadIdx.x;  // < B*C*M
  int    m  = (int)(t & (MOUT - 1));
  size_t bc = t >> 13;                                 // / MOUT
  int    c  = (int)(bc & (NCH - 1));
  int    b  = (int)(bc >> 7);
  int    id = idx[(size_t)b * MOUT + m];
  out[t] = feats[(((size_t)b * NCH + c) << 14) + id];
}

// ---------------------------------------------------------------------------
// new_xyz[b, m, k] = xyz[b, idx[b, m], k]
// ---------------------------------------------------------------------------
__global__ __launch_bounds__(256) void gather_xyz_kernel(
    const float* __restrict__ xyz, const int* __restrict__ idx,
    float* __restrict__ out) {
  int t = blockIdx.x * 256 + threadIdx.x;  // < B*M*3 = 196608 (exact grid)
  int k  = t % 3;
  int bm = t / 3;
  int m  = bm & (MOUT - 1);
  int b  = bm >> 13;
  int id = idx[(size_t)b * MOUT + m];
  out[t] = xyz[(((size_t)b << 14) + id) * 3 + k];
}

// ---------------------------------------------------------------------------
extern "C" void kernel_launch(void* const* d_in, const int* in_sizes, int n_in,
                              void* d_out, int out_size, void* d_ws, size_t ws_size,
                              hipStream_t stream) {
  (void)in_sizes; (void)n_in; (void)out_size; (void)ws_size;

  const float* feats = (const float*)d_in[0];  // (8, 128, 16384) f32
  const float* xyz   = (const float*)d_in[1];  // (8, 16384, 3)  f32

  float* out_feats = (float*)d_out;                             // (8,128,8192)
  float* out_xyz   = out_feats + (size_t)BATCH * NCH * MOUT;    // (8,8192,3)
  int*   idx       = (int*)d_ws;                                // (8,8192)

  fps_kernel<<<BATCH, TPB, 0, stream>>>(xyz, idx);

  const int feats_elems = BATCH * NCH * MOUT;   // 8,388,608
  gather_feats_kernel<<<feats_elems / 256, 256, 0, stream>>>(feats, idx, out_feats);

  const int xyz_elems = BATCH * MOUT * 3;       // 196,608 (multiple of 256)
  gather_xyz_kernel<<<xyz_elems / 256, 256, 0, stream>>>(xyz, idx, out_xyz);
}